// RankAveragePooling_64665027608601
// MI455X (gfx1250) — compile-verified
//
#include <hip/hip_runtime.h>

typedef __attribute__((ext_vector_type(2))) float v2f;
typedef __attribute__((ext_vector_type(4))) float v4f;
typedef __attribute__((ext_vector_type(8))) float v8f;
typedef __attribute__((ext_vector_type(4))) int   v4i;

#define B_SZ   32
#define D_IN   128
#define N_INST 4096
#define N_CLZ  64
#define D_RANK 25

#define W_STRIDE   132   // 132 mod 64 = 4 -> conflict-free b64 A reads across both lane halves
#define CAM_STRIDE 18    // upper-half C stores land in a disjoint bank window

// ---------------------------------------------------------------------------
// CDNA5 async-copy helpers (guarded: builtin if present, else inline asm)
// ---------------------------------------------------------------------------
template <int N>
__device__ __forceinline__ void wait_asynccnt() {
#if __has_builtin(__builtin_amdgcn_s_wait_asynccnt)
  __builtin_amdgcn_s_wait_asynccnt(N);
#else
  asm volatile("s_wait_asynccnt %0" :: "n"(N) : "memory");
#endif
  asm volatile("" ::: "memory");  // compiler barrier: no ds-load hoisting
}

__device__ __forceinline__ void async_copy_b128(const float* g, float* l) {
#if __has_builtin(__builtin_amdgcn_global_load_async_to_lds_b128)
  __builtin_amdgcn_global_load_async_to_lds_b128(
      (__attribute__((address_space(1))) v4i*)g,
      (__attribute__((address_space(3))) v4i*)l, 0, 0);
#else
  unsigned int lds_off = (unsigned int)(unsigned long long)(const void*)l;
  asm volatile("global_load_async_to_lds_b128 %0, %1, off"
               :: "v"(lds_off), "v"((unsigned long long)g)
               : "memory");
#endif
}

// ---------------------------------------------------------------------------
// Kernel A: camPe[c,n] = sum_d W[c,d]*pe[n,d]  (batch-independent, 64x4096)
// grid: N/64 blocks, 128 threads; same WMMA micro-structure as main kernel
// ---------------------------------------------------------------------------
__global__ __launch_bounds__(128) void pe_gemm_kernel(
    const float* __restrict__ W, const float* __restrict__ pe,
    float* __restrict__ camPe) {
  __shared__ float Wlds[N_CLZ * W_STRIDE];

  const int tid  = threadIdx.x;
  const int lane = tid & 31;
  const int wave = tid >> 5;
  const int half = lane >> 4;
  const int nl   = lane & 15;

  #pragma unroll
  for (int i = 0; i < 64; ++i) {
    int idx = tid + i * 128;
    Wlds[(idx >> 7) * W_STRIDE + (idx & 127)] = W[idx];
  }
  __syncthreads();

  const int n = blockIdx.x * 64 + wave * 16 + nl;

  v8f acc0 = {}, acc1 = {}, acc2 = {}, acc3 = {};
  for (int t = 0; t < 32; ++t) {
    const int kb = 4 * t + 2 * half;
    v2f bv = *(const v2f*)(pe + (size_t)n * D_IN + kb);
    v2f a0 = *(const v2f*)(&Wlds[(0 * 16 + nl) * W_STRIDE + kb]);
    v2f a1 = *(const v2f*)(&Wlds[(1 * 16 + nl) * W_STRIDE + kb]);
    v2f a2 = *(const v2f*)(&Wlds[(2 * 16 + nl) * W_STRIDE + kb]);
    v2f a3 = *(const v2f*)(&Wlds[(3 * 16 + nl) * W_STRIDE + kb]);
    acc0 = __builtin_amdgcn_wmma_f32_16x16x4_f32(false, a0, false, bv, (short)0, acc0, false, false);
    acc1 = __builtin_amdgcn_wmma_f32_16x16x4_f32(false, a1, false, bv, (short)0, acc1, false, false);
    acc2 = __builtin_amdgcn_wmma_f32_16x16x4_f32(false, a2, false, bv, (short)0, acc2, false, false);
    acc3 = __builtin_amdgcn_wmma_f32_16x16x4_f32(false, a3, false, bv, (short)0, acc3, false, false);
  }

  #pragma unroll
  for (int r = 0; r < 8; ++r) {
    const int c = r + 8 * half;
    camPe[(size_t)(0 * 16 + c) * N_INST + n] = acc0[r];
    camPe[(size_t)(1 * 16 + c) * N_INST + n] = acc1[r];
    camPe[(size_t)(2 * 16 + c) * N_INST + n] = acc2[r];
    camPe[(size_t)(3 * 16 + c) * N_INST + n] = acc3[r];
  }
}

// ---------------------------------------------------------------------------
// Kernel B: cam = W @ x per batch (fp32 WMMA, async-LDS double-buffered x),
//           epilogue adds camPe and applies the fused rank-25 mask.
// grid: B * (N/64) blocks, 128 threads (4 waves); wave: 64 classes x 16 n
// ---------------------------------------------------------------------------
__global__ __launch_bounds__(128) void cam_wmma_kernel(
    const float* __restrict__ x, const float* __restrict__ W,
    const float* __restrict__ camPe, float* __restrict__ camOut) {
  __shared__ float Wlds[N_CLZ * W_STRIDE];                  // 33792 B
  __shared__ float camS[4 * N_CLZ * CAM_STRIDE];            // 18432 B
  __shared__ __align__(16) float sx[4][2][16 * 16];         //  8192 B (per-wave dbl buf)

  const int tid  = threadIdx.x;
  const int lane = tid & 31;
  const int wave = tid >> 5;
  const int half = lane >> 4;
  const int nl   = lane & 15;

  #pragma unroll
  for (int i = 0; i < 64; ++i) {
    int idx = tid + i * 128;
    Wlds[(idx >> 7) * W_STRIDE + (idx & 127)] = W[idx];
  }
  __syncthreads();

  const int b    = blockIdx.x >> 6;
  const int nblk = blockIdx.x & 63;
  const int n0w  = nblk * 64 + wave * 16;   // this wave's 16-instance strip
  const int n    = n0w + nl;
  const float* xb = x + (size_t)b * D_IN * N_INST;

  // per-lane staging coordinates: lane covers (k-row = lane>>2, 4 n's = (lane&3)*4)
  const int srow = lane >> 2;
  const int scol = (lane & 3) * 4;
  const float* gbase = xb + (size_t)srow * N_INST + n0w + scol;  // +k*N per row block
  float* lbase = &sx[wave][0][0] + srow * 16 + scol;             // buf stride = 256 floats

  // prefetch chunk 0 (k rows 0..15) into buffer 0: 2 x b128 per wave
  async_copy_b128(gbase,                       lbase);
  async_copy_b128(gbase + (size_t)8 * N_INST,  lbase + 8 * 16);

  v8f acc0 = {}, acc1 = {}, acc2 = {}, acc3 = {};

  for (int c = 0; c < 8; ++c) {               // 8 chunks x 16 k
    if (c < 7) {                              // prefetch next chunk into other buffer
      const int k0n = (c + 1) * 16;
      float* lb = lbase + ((c + 1) & 1) * 256;
      async_copy_b128(gbase + (size_t)k0n * N_INST,       lb);
      async_copy_b128(gbase + (size_t)(k0n + 8) * N_INST, lb + 8 * 16);
      wait_asynccnt<2>();                     // chunk c's 2 loads done (in-order)
    } else {
      wait_asynccnt<0>();
    }

    const float* bufp = &sx[wave][c & 1][0];
    #pragma unroll
    for (int tl = 0; tl < 4; ++tl) {          // 4 WMMA K-steps per chunk
      const int kb = 4 * tl + 2 * half;       // lane's K pair within step
      v2f bv;
      bv.x = bufp[kb * 16 + nl];
      bv.y = bufp[(kb + 1) * 16 + nl];
      const int kg = c * 16 + kb;
      v2f a0 = *(const v2f*)(&Wlds[(0 * 16 + nl) * W_STRIDE + kg]);
      v2f a1 = *(const v2f*)(&Wlds[(1 * 16 + nl) * W_STRIDE + kg]);
      v2f a2 = *(const v2f*)(&Wlds[(2 * 16 + nl) * W_STRIDE + kg]);
      v2f a3 = *(const v2f*)(&Wlds[(3 * 16 + nl) * W_STRIDE + kg]);
      acc0 = __builtin_amdgcn_wmma_f32_16x16x4_f32(false, a0, false, bv, (short)0, acc0, false, false);
      acc1 = __builtin_amdgcn_wmma_f32_16x16x4_f32(false, a1, false, bv, (short)0, acc1, false, false);
      acc2 = __builtin_amdgcn_wmma_f32_16x16x4_f32(false, a2, false, bv, (short)0, acc2, false, false);
      acc3 = __builtin_amdgcn_wmma_f32_16x16x4_f32(false, a3, false, bv, (short)0, acc3, false, false);
    }
  }

  // stash this wave's 64x16 cam tile into per-wave LDS region
  float* cs = &camS[wave * N_CLZ * CAM_STRIDE];
  #pragma unroll
  for (int r = 0; r < 8; ++r) {
    const int c = r + 8 * half;  // C layout: VGPR r -> M = r (lanes 0-15) / 8+r (16-31)
    cs[(0 * 16 + c) * CAM_STRIDE + nl] = acc0[r];
    cs[(1 * 16 + c) * CAM_STRIDE + nl] = acc1[r];
    cs[(2 * 16 + c) * CAM_STRIDE + nl] = acc2[r];
    cs[(3 * 16 + c) * CAM_STRIDE + nl] = acc3[r];
  }

  // --- epilogue: add batch-independent camPe, then fused rank mask ---
  const float* peCol = camPe + n;           // camPe[k*N + n], coalesced across nl
  float v[64];
  #pragma unroll
  for (int k = 0; k < 64; ++k)
    v[k] = cs[k * CAM_STRIDE + nl] + peCol[(size_t)k * N_INST];

  unsigned long long keep = 0ull;
  #pragma unroll
  for (int j = 0; j < D_RANK; ++j) {
    const float vj = v[j];
    int r = 0;
    if (half == 0) {  // lower half: k = 0..31 + stable-sort tie term (k < j <= 24)
      #pragma unroll
      for (int k = 0; k < 32; ++k) r += (v[k] > vj) ? 1 : 0;
      #pragma unroll
      for (int k = 0; k < j; ++k) r += (v[k] == vj) ? 1 : 0;
    } else {          // upper half: k = 32..63
      #pragma unroll
      for (int k = 32; k < 64; ++k) r += (v[k] > vj) ? 1 : 0;
    }
    r += __shfl_xor(r, 16, 32);   // full rank of class j; both halves get it
    keep |= (1ull << r);          // position 'rank(j)' survives (idx[rank]=j < 25)
  }

  float* outb = camOut + (size_t)b * N_CLZ * N_INST + n;
  if (half == 0) {
    #pragma unroll
    for (int i = 0; i < 32; ++i)
      outb[(size_t)i * N_INST] = ((keep >> i) & 1ull) ? v[i] : 0.0f;
  } else {
    #pragma unroll
    for (int i = 0; i < 32; ++i)
      outb[(size_t)(32 + i) * N_INST] = ((keep >> (32 + i)) & 1ull) ? v[32 + i] : 0.0f;
  }
}

// ---------------------------------------------------------------------------
// Kernel C: peT[d,n] = pe[n,d]  (one-off 2MB transpose so bag reads coalesce)
// ---------------------------------------------------------------------------
__global__ void pe_transpose_kernel(const float* __restrict__ pe,
                                    float* __restrict__ peT) {
  const int idx = blockIdx.x * blockDim.x + threadIdx.x;  // d*4096 + n
  const int d = idx >> 12;
  const int nn = idx & 4095;
  peT[idx] = pe[(size_t)nn * D_IN + d];
}

// ---------------------------------------------------------------------------
// Kernel D: bag[b,d] = mean of top-25 of (x[b,d,:] + peT[d,:]); block/(b,d)
// ---------------------------------------------------------------------------
__global__ __launch_bounds__(256) void bag_kernel(const float* __restrict__ x,
                                                  const float* __restrict__ peT,
                                                  float* __restrict__ bag) {
  __shared__ float sval[256];
  __shared__ int   sidx[256];
  const int tid = threadIdx.x;
  const int b   = blockIdx.x >> 7;
  const int d   = blockIdx.x & 127;

  const float* xrow = x + ((size_t)b * D_IN + d) * N_INST;
  const float* prow = peT + (size_t)d * N_INST;
  float val[16];
  #pragma unroll
  for (int i = 0; i < 16; ++i) {
    const int nn = tid + i * 256;
    val[i] = xrow[nn] + prow[nn];
  }

  float sum = 0.0f;
  for (int it = 0; it < D_RANK; ++it) {
    float m = -__builtin_huge_valf();
    int slot = 0;
    #pragma unroll
    for (int i = 0; i < 16; ++i)
      if (val[i] > m) { m = val[i]; slot = i; }
    sval[tid] = m;
    sidx[tid] = tid;
    __syncthreads();
    for (int s = 128; s > 0; s >>= 1) {
      if (tid < s && sval[tid + s] > sval[tid]) {
        sval[tid] = sval[tid + s];
        sidx[tid] = sidx[tid + s];
      }
      __syncthreads();
    }
    const float win = sval[0];
    const int  wtid = sidx[0];
    __syncthreads();
    sum += win;
    if (tid == wtid) {
      #pragma unroll
      for (int i = 0; i < 16; ++i)
        if (i == slot) val[i] = -__builtin_huge_valf();
    }
  }
  if (tid == 0) bag[blockIdx.x] = sum * (1.0f / 25.0f);
}

// ---------------------------------------------------------------------------
// Kernel E: x passthrough — pure stream, non-temporal to keep L2 for W/pe/camPe
// ---------------------------------------------------------------------------
__global__ void copy_kernel(const v4f* __restrict__ src,
                            v4f* __restrict__ dst, int n4) {
  const int i = blockIdx.x * blockDim.x + threadIdx.x;
  if (i < n4) {
    v4f t = __builtin_nontemporal_load(&src[i]);
    __builtin_nontemporal_store(t, &dst[i]);
  }
}

// ---------------------------------------------------------------------------
// Kernel F: logits = bag @ W^T + b  (tiny; after bag)
// ---------------------------------------------------------------------------
__global__ __launch_bounds__(64) void logits_kernel(const float* __restrict__ bag,
                                                    const float* __restrict__ W,
                                                    const float* __restrict__ bias,
                                                    float* __restrict__ out) {
  const int b = blockIdx.x;
  const int c = threadIdx.x;
  const float* bg = bag + b * D_IN;
  const float* wr = W + c * D_IN;
  float s = bias[c];
  #pragma unroll
  for (int d = 0; d < D_IN; ++d) s += bg[d] * wr[d];
  out[b * N_CLZ + c] = s;
}

extern "C" void kernel_launch(void* const* d_in, const int* in_sizes, int n_in,
                              void* d_out, int out_size, void* d_ws, size_t ws_size,
                              hipStream_t stream) {
  const float* x  = (const float*)d_in[0];  // (32,128,4096)
  const float* W  = (const float*)d_in[1];  // (64,128)
  const float* bb = (const float*)d_in[2];  // (64,)
  const float* pe = (const float*)d_in[3];  // (4096,128)

  float* out    = (float*)d_out;
  float* logits = out;                               // 2048
  float* cam    = out + 2048;                        // 8388608
  float* bag    = out + 2048 + 8388608;              // 4096
  float* xout   = out + 2048 + 8388608 + 4096;       // 16777216

  float* camPe = (float*)d_ws;                       // 64*4096  = 1 MB
  float* peT   = camPe + (size_t)N_CLZ * N_INST;     // 128*4096 = 2 MB

  // precompute batch-independent pieces
  pe_gemm_kernel<<<N_INST / 64, 128, 0, stream>>>(W, pe, camPe);
  pe_transpose_kernel<<<(D_IN * N_INST) / 256, 256, 0, stream>>>(pe, peT);

  // x passthrough (offset 8394752 floats -> 16B aligned), non-temporal stream
  const int n4 = (B_SZ * D_IN * N_INST) / 4;
  copy_kernel<<<(n4 + 255) / 256, 256, 0, stream>>>((const v4f*)x, (v4f*)xout, n4);

  // bag (top-25 mean per (b,d)) using transposed pe
  bag_kernel<<<B_SZ * D_IN, 256, 0, stream>>>(x, peT, bag);

  // cam GEMM (fp32 WMMA + async-LDS x staging) + camPe add + fused rank mask
  cam_wmma_kernel<<<B_SZ * (N_INST / 64), 128, 0, stream>>>(x, W, camPe, cam);

  // logits (reads bag; same-stream ordering)
  logits_kernel<<<B_SZ, N_CLZ, 0, stream>>>(bag, W, bb, logits);
}